// ModelNew_3556232921828
// MI455X (gfx1250) — compile-verified
//
#include <hip/hip_runtime.h>
#include <cmath>

// MI455X / gfx1250, wave32. Implicit-GEMM conv via v_wmma_f32_16x16x4_f32.
// D(16oc x 16px) = A(16x4) * B(4x16) + C, K = IC*9 = 288 -> 72 WMMA / wave.
// Inner loop is pure: global_load_b64 (A) + ds_load_b64 (B) + v_wmma, all
// offsets immediate. All k-decode is hoisted to a once-per-block staging
// pass that writes B directly in WMMA fragment order into LDS.

typedef __attribute__((ext_vector_type(2))) float v2f;
typedef __attribute__((ext_vector_type(8))) float v8f;

#define IC     32
#define OC     64
#define HH     128
#define WW     128
#define NB     16
#define KTOT   288          // IC * 3 * 3
#define KSTEPS 72           // KTOT / 4
#define BELEMS (KSTEPS * 64)  // 4608 B-fragment floats per block

// ---------------------------------------------------------------------------
// Kernel 1: one-time repack into d_ws.
//  [0, 73728)        Afrag[octile(4)][j(72)][lane(32)][2 f32]  (A layout,
//                    flip + transpose folded in; ISA 16x4 f32 lane map)
//  [73728, 74880)    xoff[k=0..287] = ic*HH*WW + kh*WW + kw    (B decode LUT)
// ---------------------------------------------------------------------------
__global__ void repack_weights(const float* __restrict__ w, float* __restrict__ ws) {
  int tid = blockIdx.x * blockDim.x + threadIdx.x;
  // B-side offset LUT (straight decode; the kernel flip lives in A only).
  if (tid < KTOT) {
    int ic = tid / 9;
    int r  = tid - ic * 9;
    int kh = r / 3;
    int kw = r - kh * 3;
    ((int*)(ws + 4 * KSTEPS * 64))[tid] = ic * HH * WW + kh * WW + kw;
  }
  if (tid >= 4 * KSTEPS * 32) return;
  int lane  = tid & 31;
  int j     = (tid >> 5) % KSTEPS;
  int t     = tid / (KSTEPS * 32);
  int oc    = t * 16 + (lane & 15);
  int kbase = j * 4 + ((lane & 16) ? 2 : 0);
  float vals[2];
#pragma unroll
  for (int s = 0; s < 2; ++s) {
    int k  = kbase + s;
    int ic = k / 9;
    int r  = k - ic * 9;
    int kh = r / 3;
    int kw = r - kh * 3;
    // wflip[oc][ic][kh][kw] = weight[ic][oc][2-kh][2-kw]
    vals[s] = w[((ic * OC + oc) * 3 + (2 - kh)) * 3 + (2 - kw)];
  }
  ws[tid * 2 + 0] = vals[0];
  ws[tid * 2 + 1] = vals[1];
}

// ---------------------------------------------------------------------------
// Kernel 2: 128 threads = 4 waves; wave w -> oc tile [16w,16w+16), 16 pixels
// of row h of image n. B fragments staged in LDS in fragment order.
// ---------------------------------------------------------------------------
__global__ __launch_bounds__(128)
void convt_wmma(const float* __restrict__ x, const float* __restrict__ ws,
                const float* __restrict__ bias, float* __restrict__ out) {
  __shared__ float bfr[BELEMS];  // 18432 B: Bfrag[j][lane][2]

  const int tid = threadIdx.x;
  const int w0  = blockIdx.x * 16;
  const int h   = blockIdx.y;
  const int n   = blockIdx.z;

  const int* __restrict__ xoff = (const int*)(ws + 4 * KSTEPS * 64);
  // x base at (n, ic=0, h-1, w0-1): add xoff[k] + px for element (k, px).
  const float* __restrict__ xbase =
      x + (((size_t)n * IC) * HH + (h - 1)) * WW + (w0 - 1);

  // Block-uniform branch: interior blocks need no bounds checks at all.
  const bool interior = (h >= 1) & (h <= HH - 2) & (w0 >= 1) & (w0 + 16 <= WW - 1);

  // 4608 elements / 128 threads = 36 uniform iterations, no tail.
  // e = (j*2 + s)*32 + lane  ->  lanes 0-15 / 16-31 each read 16 consecutive
  // x floats (coalesced); LDS writes are bank-conflict-free (stride 2 dwords).
  if (interior) {
    for (int e = tid; e < BELEMS; e += 128) {
      int j = e >> 6, s = (e >> 5) & 1, lane = e & 31;
      int k = j * 4 + ((lane >> 4) << 1) + s;
      bfr[(j * 32 + lane) * 2 + s] = xbase[xoff[k] + (lane & 15)];
    }
  } else {
    for (int e = tid; e < BELEMS; e += 128) {
      int j = e >> 6, s = (e >> 5) & 1, lane = e & 31;
      int k  = j * 4 + ((lane >> 4) << 1) + s;
      int ic = k / 9;
      int r  = k - ic * 9;
      int kh = r / 3;
      int kw = r - kh * 3;
      int gh = h - 1 + kh;
      int gw = w0 - 1 + (lane & 15) + kw;
      float v = 0.0f;
      if ((unsigned)gh < (unsigned)HH && (unsigned)gw < (unsigned)WW)
        v = x[(((size_t)n * IC + ic) * HH + gh) * WW + gw];
      bfr[(j * 32 + lane) * 2 + s] = v;
    }
  }
  __syncthreads();

  const int wave = tid >> 5;
  const int lane = tid & 31;
  const int px   = lane & 15;
  const int hi   = (lane >> 4) & 1;

  // A fragments: coalesced 256 B per step, L2-hot, immediate offsets.
  const float* __restrict__ aptr = ws + (size_t)(wave * KSTEPS * 32 + lane) * 2;
  const float* __restrict__ bptr = bfr + lane * 2;

  v8f acc = {};
#pragma unroll
  for (int j = 0; j < KSTEPS; ++j) {
    v2f a = *(const v2f*)(aptr + j * 64);   // global_load_b64, offset j*256
    v2f b = *(const v2f*)(bptr + j * 64);   // ds_load_b64,    offset j*256
    acc = __builtin_amdgcn_wmma_f32_16x16x4_f32(
        /*neg_a=*/false, a, /*neg_b=*/false, b,
        /*c_mod=*/(short)0, acc, /*reuse_a=*/false, /*reuse_b=*/false);
  }

  // Epilogue. D layout: VGPR i -> M = i + 8*hi, N = px.
  const int oc_base = wave * 16 + hi * 8;
#pragma unroll
  for (int i = 0; i < 8; ++i) {
    int oc  = oc_base + i;
    float y = acc[i] + bias[oc];
    float s = __expf(y - y);                 // degenerate softmax == 1
    float r = 1.0f / (1.0f + __expf(-2.0f * s));
    out[(((size_t)n * OC + oc) * HH + h) * WW + w0 + px] = r;
  }
}

// ---------------------------------------------------------------------------
extern "C" void kernel_launch(void* const* d_in, const int* in_sizes, int n_in,
                              void* d_out, int out_size, void* d_ws,
                              size_t ws_size, hipStream_t stream) {
  const float* x    = (const float*)d_in[0];
  const float* wgt  = (const float*)d_in[1];   // (32, 64, 3, 3)
  const float* bias = (const float*)d_in[2];   // (64, 1, 1)
  float* out = (float*)d_out;
  float* ws  = (float*)d_ws;                   // 73728 B Afrag + 1152 B LUT

  const int rthreads = 4 * KSTEPS * 32;        // 9216 (also covers LUT's 288)
  repack_weights<<<(rthreads + 255) / 256, 256, 0, stream>>>(wgt, ws);

  dim3 grid(WW / 16, HH, NB);                  // (8, 128, 16)
  convt_wmma<<<grid, 128, 0, stream>>>(x, ws, bias, out);
}